// RLModel_67319317397693
// MI455X (gfx1250) — compile-verified
//
#include <hip/hip_runtime.h>
#include <stdint.h>

// ---------------- problem constants (match reference setup_inputs) ----------
#define BB 2048
#define TT 1024
#define NA 6
#define GAMMA 0.98f
#define LBDA  0.93f
#define EPSI  0.2f

// chunked segmented-scan parameters
#define CHL 64                 // time-chunk length
#define NCH (TT / CHL)         // 16 chunks per row

// policy kernel staging parameters
#define K1_IDX 512             // (b,t) indices staged per chunk
#define K1_F (K1_IDX * NA)     // 3072 floats per logits array per chunk (12KB)
#define K1_V4 (K1_F / 4)       // 768 float4 per array per chunk
#define K1_BLOCKS 512          // -> 8 chunks per block, double buffered
#define ASYNC_PER_STAGE 6      // 3 b128 per array * 2 arrays, uniform per wave

// ---------------- CDNA5 feature detection -----------------------------------
#ifndef __has_builtin
#define __has_builtin(x) 0
#endif

#if __has_builtin(__builtin_amdgcn_global_load_async_to_lds_b128)
#define HAVE_ASYNC_LDS 1
#else
#define HAVE_ASYNC_LDS 0
#endif

#if __has_builtin(__builtin_amdgcn_wmma_f32_16x16x4_f32)
#define HAVE_WMMA_F32X4 1
#else
#define HAVE_WMMA_F32X4 0
#endif

typedef __attribute__((ext_vector_type(2))) float v2f;
typedef __attribute__((ext_vector_type(8))) float v8f;
// builtin's expected pointee type per hipcc diagnostic:
typedef int v4i_vs __attribute__((vector_size(4 * sizeof(int))));
typedef __attribute__((address_space(1))) v4i_vs* g_v4i_ptr;
typedef __attribute__((address_space(3))) v4i_vs* l_v4i_ptr;

// 16B async copy global->LDS (CDNA5 GLOBAL_LOAD_ASYNC_TO_LDS_B128).
__device__ __forceinline__ void async_cp16(const float* g, float* l) {
#if HAVE_ASYNC_LDS
  __builtin_amdgcn_global_load_async_to_lds_b128(
      (g_v4i_ptr)(uintptr_t)g,
      (l_v4i_ptr)(uint32_t)(uintptr_t)l,
      /*offset=*/0, /*cpol=*/0);
#else
  *(float4*)l = *(const float4*)g;
#endif
}

template <int N>
__device__ __forceinline__ void wait_async() {
#if HAVE_ASYNC_LDS
#if __has_builtin(__builtin_amdgcn_s_wait_asynccnt)
  __builtin_amdgcn_s_wait_asynccnt((unsigned short)N);
#else
  asm volatile("s_wait_asynccnt %0" ::"n"(N) : "memory");
#endif
#endif
}

// ---------------- WMMA-based wave32 sum reduction ----------------------------
// A-operand (16x4 f32): lane L<16 holds {A[L][0],A[L][1]}, lane L>=16 holds
// {A[L-16][2],A[L-16][3]}. Put x in the first slot, 0 in the second:
//   A[m][0]=x_m, A[m][2]=x_{m+16}.  B = ones(4x16)  =>  D[m][n] = x_m + x_{m+16}.
// Lane n<16 sums VGPRs 0..7 => rows 0..7; lane n>=16 => rows 8..15.
// One cross-half shfl_xor(16) yields the wave total in every lane.
__device__ __forceinline__ float wave_sum(float x) {
#if HAVE_WMMA_F32X4
  v2f a;   a[0] = x;    a[1] = 0.0f;
  v2f one; one[0] = 1.0f; one[1] = 1.0f;
  v8f c = {0.f, 0.f, 0.f, 0.f, 0.f, 0.f, 0.f, 0.f};
  c = __builtin_amdgcn_wmma_f32_16x16x4_f32(false, a, false, one,
                                            (short)0, c, false, false);
  float s = c[0] + c[1] + c[2] + c[3] + c[4] + c[5] + c[6] + c[7];
  s += __shfl_xor(s, 16, 32);
  return s;
#else
#pragma unroll
  for (int m = 16; m > 0; m >>= 1) x += __shfl_xor(x, m, 32);
  return x;
#endif
}

// per-wave WMMA reduce -> per-block partial (fixed order => deterministic)
__device__ __forceinline__ void block_acc_store(float x, float* dst, float* sred) {
  const int w = (int)threadIdx.x >> 5;
  const float s = wave_sum(x);
  if ((threadIdx.x & 31) == 0) sred[w] = s;
  __syncthreads();
  if (threadIdx.x == 0) {
    float t = 0.f;
    const int nw = (int)blockDim.x >> 5;
    for (int i = 0; i < nw; ++i) t += sred[i];
    dst[blockIdx.x] = t;
  }
}

// chosen-action log-softmax over NA=6 contiguous floats
__device__ __forceinline__ float chosen_lse(const float* x, int a) {
  float m = x[0];
#pragma unroll
  for (int i = 1; i < NA; ++i) m = fmaxf(m, x[i]);
  float s = 0.f;
#pragma unroll
  for (int i = 0; i < NA; ++i) s += __expf(x[i] - m);
  return x[a] - m - __logf(s);
}

// ---------------- Kernel 1: policy pass (96MB of 150MB traffic) --------------
// Streams logits/logits_old through LDS with double-buffered async-to-LDS
// copies; computes ratio[b,t] and the entropy partial sums (WMMA reduce).
__device__ __forceinline__ void stage_chunk(const float* g1, const float* g2,
                                            float* l1, float* l2, int chunk) {
  const size_t gb = (size_t)chunk * K1_F;
#pragma unroll
  for (int v = 0; v < K1_V4; v += 256) {  // 3 iterations, uniform per wave
    const int i = (v + (int)threadIdx.x) * 4;
    async_cp16(g1 + gb + i, l1 + i);
    async_cp16(g2 + gb + i, l2 + i);
  }
}

__global__ __launch_bounds__(256) void k_policy(
    const float* __restrict__ logits, const float* __restrict__ logits_old,
    const int* __restrict__ actions, float* __restrict__ ratio,
    float* __restrict__ entp) {
  __shared__ __align__(16) float s_lg[2][K1_F];
  __shared__ __align__(16) float s_lo[2][K1_F];
  __shared__ float sred[8];
  const int tid = (int)threadIdx.x;
  const int per_block = (BB * TT / K1_IDX) / K1_BLOCKS;  // 8
  const int c0 = (int)blockIdx.x * per_block;
  float ent = 0.f;

  stage_chunk(logits, logits_old, s_lg[0], s_lo[0], c0);

  for (int k = 0; k < per_block; ++k) {
    const int buf = k & 1;
    if (k + 1 < per_block) {
      stage_chunk(logits, logits_old, s_lg[buf ^ 1], s_lo[buf ^ 1], c0 + k + 1);
      wait_async<ASYNC_PER_STAGE>();   // chunk k complete; k+1 may be in flight
    } else {
      wait_async<0>();
    }
    __syncthreads();
    const int base = (c0 + k) * K1_IDX;
#pragma unroll
    for (int j = 0; j < K1_IDX; j += 256) {
      const int li = j + tid;
      const int gi = base + li;
      const int a = actions[gi];
      const float lp  = chosen_lse(&s_lg[buf][li * NA], a);
      const float lpo = chosen_lse(&s_lo[buf][li * NA], a);
      ratio[gi] = __expf(lp - lpo);
      ent += lp;
    }
    __syncthreads();
  }
  block_acc_store(ent, entp, sred);
}

// ---------------- Kernel 2: per-(row,chunk) affine composition ---------------
// Both scans are x_t = a_t + b_t * x_{t+1}; done-reset folds into b_t.
// Returns:  a=r_t, b=gamma*(1-d_t)   (t==T-1: a=d?r:v, b=0)
// GAE:      a=td_t, b=gamma*lbda*(1-d_t), defined for t<=T-2.
__global__ __launch_bounds__(256) void k_composite(
    const float* __restrict__ rewards, const float* __restrict__ values,
    const unsigned char* __restrict__ dones,
    float* __restrict__ aR, float* __restrict__ bR,
    float* __restrict__ aG, float* __restrict__ bG) {
  const int id = (int)blockIdx.x * (int)blockDim.x + (int)threadIdx.x;
  if (id >= BB * NCH) return;
  const int b = id / NCH, c = id % NCH;
  const int lo = c * CHL, hi = lo + CHL;
  const float* rr = rewards + (size_t)b * TT;
  const float* vv = values + (size_t)b * TT;
  const unsigned char* dd = dones + (size_t)b * TT;
  float AR = 0.f, BR = 1.f, AG = 0.f, BG = 1.f;
  float vnext = (hi < TT) ? vv[hi] : 0.f;
  for (int t = hi - 1; t >= lo; --t) {
    const float r = rr[t], v = vv[t];
    const float nd = dd[t] ? 0.f : 1.f;
    float aRt, bRt;
    if (t == TT - 1) { aRt = dd[t] ? r : v; bRt = 0.f; }
    else             { aRt = r;             bRt = GAMMA * nd; }
    AR = aRt + bRt * AR; BR = bRt * BR;
    if (t < TT - 1) {
      const float td = r + GAMMA * nd * vnext - v;
      const float bGt = GAMMA * LBDA * nd;
      AG = td + bGt * AG; BG = bGt * BG;
    }
    vnext = v;
  }
  aR[id] = AR; bR[id] = BR; aG[id] = AG; bG[id] = BG;
}

// ---------------- Kernel 3: per-row carry scan over NCH=16 chunks ------------
__global__ __launch_bounds__(256) void k_carry(
    const float* __restrict__ aR, const float* __restrict__ bR,
    const float* __restrict__ aG, const float* __restrict__ bG,
    float* __restrict__ cR, float* __restrict__ cG) {
  const int b = (int)blockIdx.x * (int)blockDim.x + (int)threadIdx.x;
  if (b >= BB) return;
  float xR = 0.f, xG = 0.f;
  for (int c = NCH - 1; c >= 0; --c) {
    const int id = b * NCH + c;
    cR[id] = xR; cG[id] = xG;           // carry entering chunk from the right
    xR = aR[id] + bR[id] * xR;
    xG = aG[id] + bG[id] * xG;
  }
}

// ---------------- Kernel 4: replay chunks, accumulate losses -----------------
__global__ __launch_bounds__(256) void k_replay(
    const float* __restrict__ rewards, const float* __restrict__ values,
    const unsigned char* __restrict__ dones, const float* __restrict__ ratio,
    const float* __restrict__ cR, const float* __restrict__ cG,
    float* __restrict__ ppop, float* __restrict__ valp) {
  __shared__ float sred0[8];
  __shared__ float sred1[8];
  const int id = (int)blockIdx.x * (int)blockDim.x + (int)threadIdx.x;
  float psum = 0.f, vsum = 0.f;
  if (id < BB * NCH) {
    const int b = id / NCH, c = id % NCH;
    const int lo = c * CHL, hi = lo + CHL;
    const float* rr = rewards + (size_t)b * TT;
    const float* vv = values + (size_t)b * TT;
    const unsigned char* dd = dones + (size_t)b * TT;
    const float* rt = ratio + (size_t)b * TT;
    float R = cR[id], G = cG[id];
    float vnext = (hi < TT) ? vv[hi] : 0.f;
    for (int t = hi - 1; t >= lo; --t) {
      const float r = rr[t], v = vv[t];
      const float nd = dd[t] ? 0.f : 1.f;
      if (t == TT - 1) R = dd[t] ? r : v;
      else             R = r + GAMMA * nd * R;
      const float dv = R - v;
      vsum += dv * dv;
      if (t < TT - 1) {
        const float td = r + GAMMA * nd * vnext - v;
        G = td + GAMMA * LBDA * nd * G;          // = A_t
        const float ra = rt[t];
        // clip(r, 1-eps, eps) is the constant eps (faithful to reference)
        psum += fminf(ra * G, EPSI * G);
      }
      vnext = v;
    }
  }
  block_acc_store(psum, ppop, sred0);
  block_acc_store(vsum, valp, sred1);
}

// ---------------- Kernel 5: deterministic final reduction --------------------
__global__ void k_final(const float* __restrict__ entp, int ne,
                        const float* __restrict__ ppop,
                        const float* __restrict__ valp, int np,
                        float* __restrict__ out) {
  if (threadIdx.x == 0) {
    float s = 0.f;
    for (int i = 0; i < np; ++i) s += ppop[i];
    out[0] = -s / (float)((size_t)BB * (TT - 1));
  } else if (threadIdx.x == 1) {
    float s = 0.f;
    for (int i = 0; i < np; ++i) s += valp[i];
    out[1] = s / (float)((size_t)BB * TT);
  } else if (threadIdx.x == 2) {
    float s = 0.f;
    for (int i = 0; i < ne; ++i) s += entp[i];
    out[2] = -s / (float)((size_t)BB * TT);
  }
}

// ---------------- host-side launcher -----------------------------------------
extern "C" void kernel_launch(void* const* d_in, const int* in_sizes, int n_in,
                              void* d_out, int out_size, void* d_ws, size_t ws_size,
                              hipStream_t stream) {
  (void)in_sizes; (void)n_in; (void)out_size; (void)ws_size;
  const float* rewards          = (const float*)d_in[0];
  const float* values           = (const float*)d_in[1];
  const float* logits           = (const float*)d_in[2];
  const float* logits_old       = (const float*)d_in[3];
  const unsigned char* dones    = (const unsigned char*)d_in[4];  // jnp.bool_ = 1 byte
  const int* actions            = (const int*)d_in[5];
  float* out = (float*)d_out;

  // workspace layout (floats): ratio[B*T] | aR,bR,aG,bG,cR,cG[B*NCH each] |
  //                            entp[K1_BLOCKS] | ppop[NB3] | valp[NB3]
  float* w     = (float*)d_ws;
  float* ratio = w;
  float* aR    = ratio + (size_t)BB * TT;
  float* bR    = aR + (size_t)BB * NCH;
  float* aG    = bR + (size_t)BB * NCH;
  float* bG    = aG + (size_t)BB * NCH;
  float* cR    = bG + (size_t)BB * NCH;
  float* cG    = cR + (size_t)BB * NCH;
  float* entp  = cG + (size_t)BB * NCH;
  float* ppop  = entp + K1_BLOCKS;
  const int nb3 = (BB * NCH) / 256;  // 128
  float* valp  = ppop + nb3;

  k_policy<<<K1_BLOCKS, 256, 0, stream>>>(logits, logits_old, actions, ratio, entp);
  k_composite<<<(BB * NCH) / 256, 256, 0, stream>>>(rewards, values, dones,
                                                    aR, bR, aG, bG);
  k_carry<<<BB / 256, 256, 0, stream>>>(aR, bR, aG, bG, cR, cG);
  k_replay<<<(BB * NCH) / 256, 256, 0, stream>>>(rewards, values, dones, ratio,
                                                 cR, cG, ppop, valp);
  k_final<<<1, 32, 0, stream>>>(entp, K1_BLOCKS, ppop, valp, nb3, out);
}